// StackMemory_83837761618634
// MI455X (gfx1250) — compile-verified
//
#include <hip/hip_runtime.h>

#define NB 64
#define NT 1024
#define NH 512
#define NS 32

typedef __attribute__((ext_vector_type(2))) float v2f;
typedef __attribute__((ext_vector_type(8))) float v8f;
typedef __attribute__((ext_vector_type(4))) unsigned int v4u;
typedef __attribute__((ext_vector_type(8))) int v8i;
typedef __attribute__((ext_vector_type(4))) int v4i;

#define WAVES_PER_BLK 4
#define KCHUNK 64                 // K elements staged per TDM transfer
#define ROWPAD 68                 // 64 data + 4 pad DWORDs (TDM pad_enable)
#define NCHUNK (NH / KCHUNK)      // 8

// ---------------------------------------------------------------------------
// Kernel 1: P[r][0..3] = hidden[r,:] @ [Wa | Wp0 | Wp1 | Wp2]  for r in [0,B*T)
// One wave per 16-row tile. A-tiles staged global->LDS by the Tensor Data
// Mover (double buffered, TENSORcnt), with TDM row padding (4 DWORDs / 64)
// so A-fragment ds reads are bank-conflict free. GEMM via
// V_WMMA_F32_16X16X4_F32 (N padded to 16; memory-bound anyway).
// B fragments for lanes N>=4 are don't-care (their D columns are never
// stored, and WMMA columns are independent), so B loads are unpredicated.
// ---------------------------------------------------------------------------
__global__ __launch_bounds__(WAVES_PER_BLK * 32) void proj_kernel(
    const float* __restrict__ hidden, const float* __restrict__ Wa,
    const float* __restrict__ Wp, float* __restrict__ P)
{
    __shared__ float wab[NH * 4];                          // [k][n] packed
    __shared__ float stage[WAVES_PER_BLK][2][16 * ROWPAD]; // TDM landing pads

    for (int k = threadIdx.x; k < NH; k += blockDim.x) {
        wab[k * 4 + 0] = Wa[k];
        wab[k * 4 + 1] = Wp[k * 3 + 0];
        wab[k * 4 + 2] = Wp[k * 3 + 1];
        wab[k * 4 + 3] = Wp[k * 3 + 2];
    }
    __syncthreads();

    const int wave  = threadIdx.x >> 5;
    const int lane  = threadIdx.x & 31;
    const int tile  = blockIdx.x * WAVES_PER_BLK + wave;   // 16-row tile index
    const long r0   = (long)tile * 16;

    const int m     = lane & 15;   // A: row M ; B/D: col N
    const int khalf = lane >> 4;   // 0 -> K={0,1}, 1 -> K={2,3}
    const int bcol  = m & 3;       // B column actually loaded (N>=4: don't care)

    const unsigned int lds_base[2] = {
        (unsigned int)(size_t)&stage[wave][0][0],
        (unsigned int)(size_t)&stage[wave][1][0],
    };

    // ---- TDM descriptor issue: 16 rows x KCHUNK cols, row stride NH ----
    auto issue_tdm = [&](int chunk, int bufsel) {
        const unsigned long long ga =
            (unsigned long long)(hidden + r0 * NH + chunk * KCHUNK);
        v4u g0;
        g0[0] = 1u;                                   // count=1, user mode
        g0[1] = lds_base[bufsel];                     // lds_addr (bytes)
        g0[2] = (unsigned int)ga;                     // global_addr[31:0]
        g0[3] = ((unsigned int)(ga >> 32) & 0x01FFFFFFu) | (2u << 30); // type=2
        v8i g1;
        // data_size=4B(2), pad_enable, pad_interval=64 DW(5), pad_amount=4 DW(3)
        g1[0] = (2 << 16) | (1 << 20) | (5 << 22) | (3 << 25);
        g1[1] = (int)((512u & 0xFFFFu) << 16);        // tensor_dim0 = 512
        g1[2] = (int)((512u >> 16) | (4096u << 16));  // tensor_dim1 = 4096
        g1[3] = (int)((4096u >> 16) | ((unsigned)KCHUNK << 16)); // tile_dim0
        g1[4] = 16;                                   // tile_dim1=16, tile_dim2=0
        g1[5] = NH;                                   // tensor_dim0_stride = 512
        g1[6] = 0;                                    // stride0 hi | stride1 lo
        g1[7] = 0;
        const v4i gz = {0, 0, 0, 0};                  // groups 2/3 unused (2-D)
        const v8i gz8 = {0, 0, 0, 0, 0, 0, 0, 0};     // trailing arg (unused)
        __builtin_amdgcn_tensor_load_to_lds(g0, g1, gz, gz, gz8, 0);
    };

    issue_tdm(0, 0);

    v8f acc = {};
    for (int c = 0; c < NCHUNK; ++c) {
        const int cur = c & 1;
        if (c + 1 < NCHUNK) {
            issue_tdm(c + 1, cur ^ 1);
            __builtin_amdgcn_s_wait_tensorcnt(1);     // chunk c landed
        } else {
            __builtin_amdgcn_s_wait_tensorcnt(0);
        }
        const float* buf = &stage[wave][cur][0];
        const int kg0 = c * KCHUNK;                   // global K of chunk start
        #pragma unroll
        for (int kk = 0; kk < KCHUNK; kk += 4) {
            const int kl = kk + 2 * khalf;            // K pair within chunk
            const int kb = kg0 + kl;                  // global K pair
            v2f a; a.x = buf[m * ROWPAD + kl];      a.y = buf[m * ROWPAD + kl + 1];
            v2f bf; bf.x = wab[kb * 4 + bcol];      bf.y = wab[(kb + 1) * 4 + bcol];
            acc = __builtin_amdgcn_wmma_f32_16x16x4_f32(
                      false, a, false, bf, (short)0, acc, false, false);
        }
    }

    // D layout: vgpr v -> row M = v + 8*khalf, col N = lane&15
    if (m < 4) {
        const long rb = r0 + 8 * khalf;
        P[(rb + 0) * 4 + m] = acc[0];
        P[(rb + 1) * 4 + m] = acc[1];
        P[(rb + 2) * 4 + m] = acc[2];
        P[(rb + 3) * 4 + m] = acc[3];
        P[(rb + 4) * 4 + m] = acc[4];
        P[(rb + 5) * 4 + m] = acc[5];
        P[(rb + 6) * 4 + m] = acc[6];
        P[(rb + 7) * 4 + m] = acc[7];
    }
}

// ---------------------------------------------------------------------------
// Kernel 2: the sequential scan. One wave per batch element; lane s holds
// physical stack slot s as 4 scalars (da, wp0..2) + a source id. Circular
// "top" pointer makes push/pop O(1). All reductions are intra-wave shuffles.
// ids[b][logical_slot] = source index (>=0: h_t, -1: D, -2: zero).
// ---------------------------------------------------------------------------
__global__ __launch_bounds__(32) void scan_kernel(
    const float* __restrict__ P, const float* __restrict__ Wa,
    const float* __restrict__ Wp, const float* __restrict__ bp,
    const float* __restrict__ Dv, int* __restrict__ ids)
{
    const int b    = blockIdx.x;
    const int lane = threadIdx.x;

    // projections of the bottom vector D (wave-cooperative, once)
    float dDa = 0.f, dW0 = 0.f, dW1 = 0.f, dW2 = 0.f;
    for (int h = lane; h < NH; h += 32) {
        const float d = Dv[h];
        dDa += d * Wa[h];
        dW0 += d * Wp[h * 3 + 0];
        dW1 += d * Wp[h * 3 + 1];
        dW2 += d * Wp[h * 3 + 2];
    }
    #pragma unroll
    for (int off = 16; off >= 1; off >>= 1) {
        dDa += __shfl_xor(dDa, off, 32);
        dW0 += __shfl_xor(dW0, off, 32);
        dW1 += __shfl_xor(dW1, off, 32);
        dW2 += __shfl_xor(dW2, off, 32);
    }
    const float bp0 = bp[0], bp1 = bp[1], bp2 = bp[2];
    // note: ba shifts all softmax inputs equally -> softmax invariant, skip it

    float sDa = 0.f, sW0 = 0.f, sW1 = 0.f, sW2 = 0.f;  // slot projections
    int   sId = -2;                                    // -2 == zero vector
    int   top = 0;                                     // lane-uniform

    const float4* Pb = (const float4*)P + (long)b * NT;
    for (int t = 0; t < NT; ++t) {
        const float4 p = Pb[t];   // {da_t, hp0, hp1, hp2} (broadcast load)

        // softmax over the 32 slot-dots (max-sub for stability, like jax)
        float mx = sDa;
        #pragma unroll
        for (int off = 16; off >= 1; off >>= 1)
            mx = fmaxf(mx, __shfl_xor(mx, off, 32));
        const float e = __expf(sDa - mx);

        // fused: Z = sum e ; Sj = sum e*wp_j  ->  context@Wp = Sj / Z
        float r0 = e, r1 = e * sW0, r2 = e * sW1, r3 = e * sW2;
        #pragma unroll
        for (int off = 16; off >= 1; off >>= 1) {
            r0 += __shfl_xor(r0, off, 32);
            r1 += __shfl_xor(r1, off, 32);
            r2 += __shfl_xor(r2, off, 32);
            r3 += __shfl_xor(r3, off, 32);
        }
        const float inv = 1.0f / r0;
        const float l0 = p.y + r1 * inv + bp0;
        const float l1 = p.z + r2 * inv + bp1;
        const float l2 = p.w + r3 * inv + bp2;

        int a = 0; float best = l0;              // first-max tie rule
        if (l1 > best) { best = l1; a = 1; }
        if (l2 > best) { a = 2; }

        if (a == 0) {                             // push: ht on top
            top = (top - 1) & 31;
            if (lane == top) { sDa = p.x; sW0 = p.y; sW1 = p.z; sW2 = p.w; sId = t; }
        } else if (a == 1) {                      // pop: D enters at bottom
            if (lane == top) { sDa = dDa; sW0 = dW0; sW1 = dW1; sW2 = dW2; sId = -1; }
            top = (top + 1) & 31;
        }
    }
    ids[b * NS + ((lane - top) & 31)] = sId;
}

// ---------------------------------------------------------------------------
// Kernel 3: materialize final_stack[b][s][:] from the recorded source ids.
// ---------------------------------------------------------------------------
__global__ __launch_bounds__(128) void gather_kernel(
    const float* __restrict__ hidden, const float* __restrict__ Dv,
    const int* __restrict__ ids, float* __restrict__ out)
{
    const int bs = blockIdx.x;          // b*NS + s
    const int b  = bs >> 5;
    const int id = ids[bs];
    const int i  = threadIdx.x;         // 128 threads x float4 = 512 floats

    float4 v;
    if (id >= 0)       v = ((const float4*)(hidden + ((long)b * NT + id) * NH))[i];
    else if (id == -1) v = ((const float4*)Dv)[i];
    else               v = make_float4(0.f, 0.f, 0.f, 0.f);

    ((float4*)(out + (long)bs * NH))[i] = v;
}

// ---------------------------------------------------------------------------
extern "C" void kernel_launch(void* const* d_in, const int* in_sizes, int n_in,
                              void* d_out, int out_size, void* d_ws, size_t ws_size,
                              hipStream_t stream) {
    const float* hidden = (const float*)d_in[0];   // (B,T,H) f32
    const float* Wa     = (const float*)d_in[1];   // (H,1)
    // d_in[2] = ba (softmax shift-invariant, unused)
    const float* Wp     = (const float*)d_in[3];   // (H,3)
    const float* bp     = (const float*)d_in[4];   // (3,)
    const float* Dv     = (const float*)d_in[5];   // (1,H)
    float*       out    = (float*)d_out;           // (B,S,H)

    float* P   = (float*)d_ws;                                              // B*T*4 f32
    int*   ids = (int*)((char*)d_ws + (size_t)NB * NT * 4 * sizeof(float)); // B*S i32

    proj_kernel  <<<(NB * NT / 16) / WAVES_PER_BLK, WAVES_PER_BLK * 32, 0, stream>>>(
        hidden, Wa, Wp, P);
    scan_kernel  <<<NB,       32, 0, stream>>>(P, Wa, Wp, bp, Dv, ids);
    gather_kernel<<<NB * NS, 128, 0, stream>>>(hidden, Dv, ids, out);
}